// GAT_6064493822274
// MI455X (gfx1250) — compile-verified
//
#include <hip/hip_runtime.h>
#include <math.h>

// ---------------- problem constants ----------------
#define N_NODES   50000
#define N_EDGES   400000
#define E_TOT     450000          // edges + self loops
#define DIM_IN    768
#define DIM_H     32
#define HEADS     8
#define HID       256             // HEADS*DIM_H
#define DIM_OUT   16
#define NEG_SLOPE 0.2f
#define OUT_HALF  (N_NODES * DIM_OUT)   // 800000

#define MT        5               // 16-row M-tiles per block (80 rows); 50000 = 625*80
#define LDS_STR   68              // padded row stride (floats) -> conflict-free A reads

typedef __attribute__((ext_vector_type(2))) float v2f;
typedef __attribute__((ext_vector_type(8))) float v8f;

// monotone float <-> uint mapping so atomicMax(u32) implements float max
__device__ __forceinline__ unsigned enc_f(float f) {
  unsigned u = __float_as_uint(f);
  return (u & 0x80000000u) ? ~u : (u | 0x80000000u);
}
__device__ __forceinline__ float dec_f(unsigned u) {
  return __uint_as_float((u & 0x80000000u) ? (u ^ 0x80000000u) : ~u);
}
#define ENC_NEG_INF 0x007FFFFFu   // enc_f(-inf)

__device__ __forceinline__ float lrelu(float v) {
  return v > 0.f ? v : v * NEG_SLOPE;
}

__device__ __forceinline__ v8f wmma4(v2f a, v2f b, v8f c) {
  return __builtin_amdgcn_wmma_f32_16x16x4_f32(false, a, false, b,
                                               (short)0, c, false, false);
}

// ---------------------------------------------------------------------------
// Layer-1 GEMM:  C[50000 x 256] = X[50000 x 768] @ W[768 x 256]   (f32 WMMA)
// grid = 625 blocks of 256 threads (8 waves). Block computes an 80-row stripe
// staged through LDS (padded stride 68). Wave w owns column tiles w and w+8
// (16 tiles total), with 5 m-tiles each -> 10 accumulators, so each B-fragment
// pair feeds 5 WMMAs and each A-fragment feeds 2.
// A frag layout (ISA 7.12.2, 32-bit 16x4): lane<16 -> K=0,1 ; lane>=16 -> K=2,3.
// ---------------------------------------------------------------------------
__global__ __launch_bounds__(256)
void gemm1_wmma(const float* __restrict__ X, const float* __restrict__ W,
                float* __restrict__ C) {
  __shared__ float lds[MT * 16 * LDS_STR];
  const int tid  = threadIdx.x;
  const int wave = __builtin_amdgcn_readfirstlane(tid >> 5);   // SGPR-uniform
  const int lane = tid & 31;
  const int m0   = blockIdx.x * (16 * MT);
  const int mrow = lane & 15;
  const int kk0  = (lane >> 4) << 1;                           // 0 or 2
  const int nt0  = wave, nt1 = wave + 8;

  v8f acc0[MT], acc1[MT];
  #pragma unroll
  for (int mt = 0; mt < MT; ++mt) { acc0[mt] = (v8f){}; acc1[mt] = (v8f){}; }

  const int ldr = tid >> 4;            // 0..15 (staging row within sub-tile)
  const int ldc = (tid & 15) << 2;     // 0..60 (float4 column)

  for (int kc = 0; kc < DIM_IN; kc += 64) {
    __syncthreads();
    #pragma unroll
    for (int p = 0; p < MT; ++p) {
      const int row = ldr + p * 16;
      float4 v = *(const float4*)(X + (size_t)(m0 + row) * DIM_IN + kc + ldc);
      *(float4*)(&lds[row * LDS_STR + ldc]) = v;
    }
    if (kc + 64 < DIM_IN)   // prefetch next chunk (global_prefetch_b8)
      __builtin_prefetch(X + (size_t)(m0 + ldr) * DIM_IN + kc + 64 + ldc, 0, 1);
    __syncthreads();

    #pragma unroll 4
    for (int k2 = 0; k2 < 64; k2 += 4) {
      const int kg = kc + k2 + kk0;
      v2f b0, b1;
      b0.x = W[kg * HID + nt0 * 16 + mrow];
      b0.y = W[(kg + 1) * HID + nt0 * 16 + mrow];
      b1.x = W[kg * HID + nt1 * 16 + mrow];
      b1.y = W[(kg + 1) * HID + nt1 * 16 + mrow];
      #pragma unroll
      for (int mt = 0; mt < MT; ++mt) {
        v2f a;
        a.x = lds[(mt * 16 + mrow) * LDS_STR + k2 + kk0];
        a.y = lds[(mt * 16 + mrow) * LDS_STR + k2 + kk0 + 1];
        acc0[mt] = wmma4(a, b0, acc0[mt]);
        acc1[mt] = wmma4(a, b1, acc1[mt]);
      }
    }
  }
  // C/D layout: VGPR r -> row +r (lanes<16) / +8+r (lanes>=16), col = nt*16 + (lane&15)
  const int mhalf = (lane >> 4) << 3;
  const int n0 = nt0 * 16 + mrow, n1 = nt1 * 16 + mrow;
  #pragma unroll
  for (int mt = 0; mt < MT; ++mt) {
    const int rb = m0 + mt * 16 + mhalf;
    #pragma unroll
    for (int r = 0; r < 8; ++r) {
      C[(size_t)(rb + r) * HID + n0] = acc0[mt][r];
      C[(size_t)(rb + r) * HID + n1] = acc1[mt][r];
    }
  }
}

// ---------------------------------------------------------------------------
// Layer-2 dual GEMM: OL = H @ Wl, OR = H @ Wr  with H[50000 x 256], W[256 x 16].
// Same 80-row stripe per block; 10 tasks = (5 m-tiles) x (2 weights) spread
// over 8 waves (waves 0,1 carry a second task), so no wave idles.
// ---------------------------------------------------------------------------
__global__ __launch_bounds__(256)
void gemm2_wmma(const float* __restrict__ H, const float* __restrict__ Wl,
                const float* __restrict__ Wr, float* __restrict__ OL,
                float* __restrict__ OR_) {
  __shared__ float lds[MT * 16 * LDS_STR];
  const int tid  = threadIdx.x;
  const int wave = __builtin_amdgcn_readfirstlane(tid >> 5);
  const int lane = tid & 31;
  const int m0   = blockIdx.x * (16 * MT);
  const int mrow = lane & 15;
  const int kk0  = (lane >> 4) << 1;

  const int  mt0  = wave % MT;          // task 0: tile mt0 of weight sel0
  const int  sel0 = wave / MT;
  const bool has1 = (wave < 2);         // task 1 (tasks 8,9): tiles 3,4 of Wr
  const int  mt1  = wave + 3;
  const float* W0 = sel0 ? Wr : Wl;
  float*       O0 = sel0 ? OR_ : OL;

  v8f acc0 = {}, acc1 = {};
  const int ldr = tid >> 4;
  const int ldc = (tid & 15) << 2;

  for (int kc = 0; kc < HID; kc += 64) {
    __syncthreads();
    #pragma unroll
    for (int p = 0; p < MT; ++p) {
      const int row = ldr + p * 16;
      float4 v = *(const float4*)(H + (size_t)(m0 + row) * HID + kc + ldc);
      *(float4*)(&lds[row * LDS_STR + ldc]) = v;
    }
    __syncthreads();

    #pragma unroll 4
    for (int k2 = 0; k2 < 64; k2 += 4) {
      const int kg = kc + k2 + kk0;
      v2f a0, b0;
      a0.x = lds[(mt0 * 16 + mrow) * LDS_STR + k2 + kk0];
      a0.y = lds[(mt0 * 16 + mrow) * LDS_STR + k2 + kk0 + 1];
      b0.x = W0[kg * DIM_OUT + mrow];
      b0.y = W0[(kg + 1) * DIM_OUT + mrow];
      acc0 = wmma4(a0, b0, acc0);
      if (has1) {                        // wave-uniform scalar branch
        v2f a1, b1;
        a1.x = lds[(mt1 * 16 + mrow) * LDS_STR + k2 + kk0];
        a1.y = lds[(mt1 * 16 + mrow) * LDS_STR + k2 + kk0 + 1];
        b1.x = Wr[kg * DIM_OUT + mrow];
        b1.y = Wr[(kg + 1) * DIM_OUT + mrow];
        acc1 = wmma4(a1, b1, acc1);
      }
    }
  }
  const int mhalf = (lane >> 4) << 3;
  {
    const int rb = m0 + mt0 * 16 + mhalf;
    #pragma unroll
    for (int r = 0; r < 8; ++r) O0[(size_t)(rb + r) * DIM_OUT + mrow] = acc0[r];
  }
  if (has1) {
    const int rb = m0 + mt1 * 16 + mhalf;
    #pragma unroll
    for (int r = 0; r < 8; ++r) OR_[(size_t)(rb + r) * DIM_OUT + mrow] = acc1[r];
  }
}

// ---------------------------------------------------------------------------
__global__ void init_kernel(unsigned* M1, float* D1, unsigned* M2, float* D2,
                            float* out) {
  int i = blockIdx.x * blockDim.x + threadIdx.x;
  if (i < N_NODES * HEADS) { M1[i] = ENC_NEG_INF; D1[i] = 0.f; }
  if (i < N_NODES)         { M2[i] = ENC_NEG_INF; D2[i] = 0.f; }
  if (i < OUT_HALF)        out[i] = 0.f;
}

__global__ void zero_f4(float4* p, int n4) {
  int i = blockIdx.x * blockDim.x + threadIdx.x;
  if (i < n4) p[i] = make_float4(0.f, 0.f, 0.f, 0.f);
}

__device__ __forceinline__ void edge_sd(const int* ei, int e, int& src, int& dst) {
  if (e < N_EDGES) { src = ei[e]; dst = ei[N_EDGES + e]; }
  else             { src = dst = e - N_EDGES; }
}

// ---- layer 1 edge phase (8 heads x 32 ch) ----
__global__ void score1_kernel(const float* __restrict__ xl, const float* __restrict__ xr,
                              const int* __restrict__ ei, const float* __restrict__ att,
                              float* __restrict__ S, unsigned* __restrict__ Mx) {
  int t = blockIdx.x * blockDim.x + threadIdx.x;
  if (t >= E_TOT * HEADS) return;
  int e = t >> 3, h = t & 7, src, dst;
  edge_sd(ei, e, src, dst);
  const float4* pl = (const float4*)(xl + (size_t)src * HID + h * DIM_H);
  const float4* pr = (const float4*)(xr + (size_t)dst * HID + h * DIM_H);
  const float4* pa = (const float4*)(att + h * DIM_H);
  float s = 0.f;
  #pragma unroll
  for (int q = 0; q < 8; ++q) {
    float4 a = pl[q], b = pr[q], w = pa[q];
    s += w.x * lrelu(a.x + b.x);
    s += w.y * lrelu(a.y + b.y);
    s += w.z * lrelu(a.z + b.z);
    s += w.w * lrelu(a.w + b.w);
  }
  S[t] = s;
  atomicMax(&Mx[(size_t)dst * HEADS + h], enc_f(s));
}

__global__ void exp1_kernel(const int* __restrict__ ei, float* __restrict__ S,
                            const unsigned* __restrict__ Mx, float* __restrict__ D) {
  int t = blockIdx.x * blockDim.x + threadIdx.x;
  if (t >= E_TOT * HEADS) return;
  int e = t >> 3, h = t & 7, src, dst;
  edge_sd(ei, e, src, dst);
  float m  = dec_f(Mx[(size_t)dst * HEADS + h]);
  float ex = expf(S[t] - m);
  S[t] = ex;
  atomicAdd(&D[(size_t)dst * HEADS + h], ex);
}

__global__ void agg1_kernel(const int* __restrict__ ei, const float* __restrict__ S,
                            const float* __restrict__ D, const float* __restrict__ xl,
                            float* __restrict__ out) {
  int t = blockIdx.x * blockDim.x + threadIdx.x;
  if (t >= E_TOT * HEADS) return;
  int e = t >> 3, h = t & 7, src, dst;
  edge_sd(ei, e, src, dst);
  float alpha = S[t] / D[(size_t)dst * HEADS + h];
  const float* a = xl + (size_t)src * HID + h * DIM_H;
  float*       o = out + (size_t)dst * HID + h * DIM_H;
  #pragma unroll
  for (int c = 0; c < DIM_H; ++c) atomicAdd(&o[c], alpha * a[c]);
}

__global__ void biaselu_kernel(float* __restrict__ h, const float* __restrict__ b) {
  int i = blockIdx.x * blockDim.x + threadIdx.x;
  if (i >= N_NODES * HID) return;
  float v = h[i] + b[i & (HID - 1)];
  h[i] = v > 0.f ? v : (expf(v) - 1.f);
}

// ---- layer 2 edge phase (1 head x 16 ch) ----
__global__ void score2_kernel(const float* __restrict__ xl, const float* __restrict__ xr,
                              const int* __restrict__ ei, const float* __restrict__ att,
                              float* __restrict__ S, unsigned* __restrict__ Mx) {
  int e = blockIdx.x * blockDim.x + threadIdx.x;
  if (e >= E_TOT) return;
  int src, dst;
  edge_sd(ei, e, src, dst);
  const float4* pl = (const float4*)(xl + (size_t)src * DIM_OUT);
  const float4* pr = (const float4*)(xr + (size_t)dst * DIM_OUT);
  const float4* pa = (const float4*)att;
  float s = 0.f;
  #pragma unroll
  for (int q = 0; q < 4; ++q) {
    float4 a = pl[q], b = pr[q], w = pa[q];
    s += w.x * lrelu(a.x + b.x);
    s += w.y * lrelu(a.y + b.y);
    s += w.z * lrelu(a.z + b.z);
    s += w.w * lrelu(a.w + b.w);
  }
  S[e] = s;
  atomicMax(&Mx[dst], enc_f(s));
}

__global__ void exp2_kernel(const int* __restrict__ ei, float* __restrict__ S,
                            const unsigned* __restrict__ Mx, float* __restrict__ D) {
  int e = blockIdx.x * blockDim.x + threadIdx.x;
  if (e >= E_TOT) return;
  int src, dst;
  edge_sd(ei, e, src, dst);
  float ex = expf(S[e] - dec_f(Mx[dst]));
  S[e] = ex;
  atomicAdd(&D[dst], ex);
}

__global__ void agg2_kernel(const int* __restrict__ ei, const float* __restrict__ S,
                            const float* __restrict__ D, const float* __restrict__ xl,
                            float* __restrict__ out) {
  int e = blockIdx.x * blockDim.x + threadIdx.x;
  if (e >= E_TOT) return;
  int src, dst;
  edge_sd(ei, e, src, dst);
  float alpha = S[e] / D[dst];
  const float* a = xl + (size_t)src * DIM_OUT;
  float*       o = out + (size_t)dst * DIM_OUT;
  #pragma unroll
  for (int c = 0; c < DIM_OUT; ++c) atomicAdd(&o[c], alpha * a[c]);
}

__global__ void finalize_kernel(float* __restrict__ out, const float* __restrict__ b2) {
  int i = blockIdx.x * blockDim.x + threadIdx.x;
  if (i >= N_NODES) return;
  float v[DIM_OUT];
  float mx = -INFINITY;
  #pragma unroll
  for (int c = 0; c < DIM_OUT; ++c) {
    v[c] = out[(size_t)i * DIM_OUT + c] + b2[c];
    mx = fmaxf(mx, v[c]);
  }
  float se = 0.f;
  #pragma unroll
  for (int c = 0; c < DIM_OUT; ++c) se += expf(v[c] - mx);
  float lse = mx + logf(se);
  #pragma unroll
  for (int c = 0; c < DIM_OUT; ++c) {
    out[(size_t)i * DIM_OUT + c] = v[c];
    out[OUT_HALF + (size_t)i * DIM_OUT + c] = v[c] - lse;
  }
}

// ---------------------------------------------------------------------------
extern "C" void kernel_launch(void* const* d_in, const int* in_sizes, int n_in,
                              void* d_out, int out_size, void* d_ws, size_t ws_size,
                              hipStream_t stream) {
  const float* x    = (const float*)d_in[0];
  const int*   ei   = (const int*)  d_in[1];
  const float* W1l  = (const float*)d_in[2];
  const float* W1r  = (const float*)d_in[3];
  const float* att1 = (const float*)d_in[4];
  const float* b1   = (const float*)d_in[5];
  const float* W2l  = (const float*)d_in[6];
  const float* W2r  = (const float*)d_in[7];
  const float* att2 = (const float*)d_in[8];
  const float* b2   = (const float*)d_in[9];
  float* out = (float*)d_out;

  // workspace layout (floats) — total 30,550,000 floats = 122.2 MB
  float* ws = (float*)d_ws;
  float*    A   = ws;                          // xl1 [N,256]; later xl2/xr2
  float*    B   = ws + 12800000;               // xr1 -> agg1 target -> h1
  float*    S1  = ws + 25600000;               // [E_TOT*8]
  unsigned* M1  = (unsigned*)(ws + 29200000);  // [N*8]
  float*    D1  = ws + 29600000;               // [N*8]
  float*    S2  = ws + 30000000;               // [E_TOT]
  unsigned* M2  = (unsigned*)(ws + 30450000);  // [N]
  float*    D2  = ws + 30500000;               // [N]
  float*    XL2 = A;                           // [N,16]
  float*    XR2 = A + OUT_HALF;                // [N,16]

  const int BLK = 256;
  const int gBlocks = N_NODES / (16 * MT);     // 625

  init_kernel<<<(OUT_HALF + BLK - 1) / BLK, BLK, 0, stream>>>(M1, D1, M2, D2, out);

  // layer 1 projections (f32 WMMA)
  gemm1_wmma<<<gBlocks, BLK, 0, stream>>>(x, W1l, A);
  gemm1_wmma<<<gBlocks, BLK, 0, stream>>>(x, W1r, B);

  // layer 1 attention
  int t1 = E_TOT * HEADS;
  score1_kernel<<<(t1 + BLK - 1) / BLK, BLK, 0, stream>>>(A, B, ei, att1, S1, M1);
  zero_f4<<<(N_NODES * HID / 4 + BLK - 1) / BLK, BLK, 0, stream>>>((float4*)B,
                                                                  N_NODES * HID / 4);
  exp1_kernel<<<(t1 + BLK - 1) / BLK, BLK, 0, stream>>>(ei, S1, M1, D1);
  agg1_kernel<<<(t1 + BLK - 1) / BLK, BLK, 0, stream>>>(ei, S1, D1, A, B);
  biaselu_kernel<<<(N_NODES * HID + BLK - 1) / BLK, BLK, 0, stream>>>(B, b1);

  // layer 2 projections (both weights in one kernel)
  gemm2_wmma<<<gBlocks, BLK, 0, stream>>>(B, W2l, W2r, XL2, XR2);

  // layer 2 attention -> accumulate into d_out[0:OUT_HALF]
  score2_kernel<<<(E_TOT + BLK - 1) / BLK, BLK, 0, stream>>>(XL2, XR2, ei, att2, S2, M2);
  exp2_kernel<<<(E_TOT + BLK - 1) / BLK, BLK, 0, stream>>>(ei, S2, M2, D2);
  agg2_kernel<<<(E_TOT + BLK - 1) / BLK, BLK, 0, stream>>>(ei, S2, D2, XL2, out);

  finalize_kernel<<<(N_NODES + BLK - 1) / BLK, BLK, 0, stream>>>(out, b2);
}